// DASOFeatureQueue_10325101380108
// MI455X (gfx1250) — compile-verified
//
#include <hip/hip_runtime.h>
#include <hip/hip_bf16.h>

#define C_    1000
#define Q_    256
#define D_    128
#define B_    65536
#define CHUNK 256
#define NCH   (B_ / CHUNK)   // 256 chunks

typedef __attribute__((ext_vector_type(2))) float v2f;
typedef __attribute__((ext_vector_type(8))) float v8f;

// ---------------- init slot_owner to -1 ----------------
__global__ void k_init_owner(int* owner) {
    int i = blockIdx.x * blockDim.x + threadIdx.x;
    if (i < C_ * Q_) owner[i] = -1;
}

// ---------------- per-chunk class histogram ----------------
__global__ void k_hist(const int* __restrict__ labels, int* __restrict__ H) {
    __shared__ int h[C_];
    int ch = blockIdx.x;
    for (int c = threadIdx.x; c < C_; c += blockDim.x) h[c] = 0;
    __syncthreads();
    int lab = labels[ch * CHUNK + threadIdx.x];
    atomicAdd(&h[lab], 1);
    __syncthreads();
    for (int c = threadIdx.x; c < C_; c += blockDim.x) H[ch * C_ + c] = h[c];
}

// ---------------- per-class scan over chunks: offsets, n_c, cnt_new, valid ----------------
__global__ void k_scan(const int* __restrict__ H, int* __restrict__ O,
                       int* __restrict__ n_c, const int* __restrict__ cnt,
                       int* __restrict__ cnt_new, float* __restrict__ valid_out) {
    int c = blockIdx.x * blockDim.x + threadIdx.x;
    if (c >= C_) return;
    int run = 0;
    for (int ch = 0; ch < NCH; ++ch) {   // coalesced across threads per iteration
        O[ch * C_ + c] = run;
        run += H[ch * C_ + c];
    }
    n_c[c] = run;
    int cn = cnt[c] + run;
    if (cn > Q_) cn = Q_;
    cnt_new[c] = cn;
    valid_out[c] = (cn > 0) ? 1.0f : 0.0f;
}

// ---------------- stable in-chunk rank + deterministic scatter of slot owners ----------------
__global__ void k_rank(const int* __restrict__ labels, const int* __restrict__ O,
                       const int* __restrict__ n_c, const int* __restrict__ ptrv,
                       int* __restrict__ owner) {
    __shared__ int sl[CHUNK];
    int ch = blockIdx.x, i = threadIdx.x;
    int lab = labels[ch * CHUNK + i];
    sl[i] = lab;
    __syncthreads();
    int r = 0;
    for (int j = 0; j < i; ++j) r += (sl[j] == lab) ? 1 : 0;  // batch-order stable rank
    int rank = O[ch * C_ + lab] + r;
    int n = n_c[lab];
    if (rank >= n - Q_) {                      // torch keeps only last Q per class
        int slot = (ptrv[lab] + rank) % Q_;    // ring-buffer position; unique per (class,slot)
        owner[lab * Q_ + slot] = ch * CHUNK + i;
    }
}

// ---------------- per-row inverse L2 norm (one wave32 per row) ----------------
__global__ void k_norm(const float* __restrict__ feats, float* __restrict__ invnorm) {
    int lane = threadIdx.x & 31;
    int w = threadIdx.x >> 5;
    int row = blockIdx.x * 8 + w;
    const float4* s4 = (const float4*)(feats + (size_t)row * D_);
    float4 v = s4[lane];                               // 32 lanes x float4 = 128 floats
    float s = v.x * v.x + v.y * v.y + v.z * v.z + v.w * v.w;
    for (int m = 16; m; m >>= 1) s += __shfl_xor(s, m, 32);
    if (lane == 0) invnorm[row] = 1.0f / fmaxf(sqrtf(s), 1e-12f);
}

// ---------------- per-class masked, scaled mean via V_WMMA_F32_16X16X4_F32, then l2norm ----
// Scale (invnorm for kept rows, 1.0 for bank rows) is folded into the WMMA A operand:
// A[m][k] = mask(q) * scale(q); B = raw staged rows.
__global__ void __launch_bounds__(32)
k_proto(const float* __restrict__ feats, const float* __restrict__ bank,
        const float* __restrict__ invnorm, const int* __restrict__ owner,
        const int* __restrict__ cnt_new, float* __restrict__ out) {
    __shared__ float sm[16 * D_];      // 16 staged q-rows x 128 dims = 8KB
    int c = blockIdx.x;
    int lane = threadIdx.x;            // full wave32, EXEC all-1s for WMMA
    int cntv = cnt_new[c];

    v8f acc[8] = {};                   // 8 N-tiles cover D=128
    int nn  = lane & 15;               // N within tile
    int khi = (lane >> 4) << 1;        // K base per half-wave: 0 or 2

    for (int q0 = 0; q0 < cntv; q0 += 16) {
        // lanes (mod 16) fetch this chunk's 16 owners + per-row scales up front
        int qr = q0 + nn;
        int own = (qr < Q_) ? owner[c * Q_ + qr] : -1;
        float scv = 0.0f;
        if (qr < cntv) scv = (own >= 0) ? invnorm[own] : 1.0f;   // mask*scale
        if (q0 + 16 < cntv)                                      // global_prefetch_b8
            __builtin_prefetch(&owner[c * Q_ + q0 + 16 + nn], 0, 0);

        // stage 16 raw rows; owner broadcast via shfl so row loads pipeline
        #pragma unroll 4
        for (int r = 0; r < 16; ++r) {
            int q = q0 + r;
            int own_r = __shfl(own, r, 32);        // wave-uniform
            float4 v;
            if (q < Q_) {
                const float4* s4 = (own_r >= 0)
                    ? (const float4*)(feats + (size_t)own_r * D_)
                    : (const float4*)(bank + ((size_t)c * Q_ + q) * D_);
                v = s4[lane];                      // 512B coalesced row copy
            } else {
                v = make_float4(0.f, 0.f, 0.f, 0.f);
            }
            ((float4*)sm)[r * 32 + lane] = v;
        }
        __syncthreads();

        #pragma unroll
        for (int k4 = 0; k4 < 4; ++k4) {
            // A = (mask*scale) broadcast over M, K-striped per §7.12.2
            v2f a;
            a.x = __shfl(scv, 4 * k4 + khi,     32);   // VGPR0: K = khi
            a.y = __shfl(scv, 4 * k4 + khi + 1, 32);   // VGPR1: K = khi+1
            int rb = (k4 * 4 + khi) * D_ + nn;
            #pragma unroll
            for (int t = 0; t < 8; ++t) {
                v2f b;                                 // B[k][n] tile from LDS
                b.x = sm[rb        + t * 16];
                b.y = sm[rb + D_   + t * 16];
                acc[t] = __builtin_amdgcn_wmma_f32_16x16x4_f32(
                    false, a, false, b, (short)0, acc[t], false, false);
            }
        }
        __syncthreads();
    }

    // D rows are all identical (A rows identical): row M=0 lives in acc[t][0], lane = N
    float invc = 1.0f / (float)(cntv > 0 ? cntv : 1);
    float m[8];
    float nsq = 0.f;
    #pragma unroll
    for (int t = 0; t < 8; ++t) { m[t] = acc[t][0] * invc; nsq += m[t] * m[t]; }
    for (int mm = 8; mm; mm >>= 1) nsq += __shfl_xor(nsq, mm, 16);  // reduce within 16-lane half
    float invn = 1.0f / fmaxf(sqrtf(nsq), 1e-12f);
    if (lane < 16) {
        #pragma unroll
        for (int t = 0; t < 8; ++t)
            out[(size_t)c * D_ + t * 16 + nn] = m[t] * invn;
    }
}

extern "C" void kernel_launch(void* const* d_in, const int* in_sizes, int n_in,
                              void* d_out, int out_size, void* d_ws, size_t ws_size,
                              hipStream_t stream) {
    const float* feats  = (const float*)d_in[0];
    const float* bank   = (const float*)d_in[1];
    const int*   labels = (const int*)d_in[2];
    const int*   ptrv   = (const int*)d_in[3];
    const int*   cnt    = (const int*)d_in[4];

    float* proto = (float*)d_out;                  // [C,D]
    float* valid = proto + (size_t)C_ * D_;        // [C] as 0/1 floats

    char* w = (char*)d_ws;
    float* invnorm = (float*)w;  w += (size_t)B_ * 4;          // 256 KB
    int*   owner   = (int*)w;    w += (size_t)C_ * Q_ * 4;     // 1 MB
    int*   H       = (int*)w;    w += (size_t)NCH * C_ * 4;    // 1 MB
    int*   O       = (int*)w;    w += (size_t)NCH * C_ * 4;    // 1 MB
    int*   n_c     = (int*)w;    w += 4096;
    int*   cnt_new = (int*)w;    w += 4096;

    k_init_owner<<<(C_ * Q_ + 255) / 256, 256, 0, stream>>>(owner);
    k_hist <<<NCH, CHUNK, 0, stream>>>(labels, H);
    k_scan <<<(C_ + 255) / 256, 256, 0, stream>>>(H, O, n_c, cnt, cnt_new, valid);
    k_rank <<<NCH, CHUNK, 0, stream>>>(labels, O, n_c, ptrv, owner);
    k_norm <<<B_ / 8, 256, 0, stream>>>(feats, invnorm);
    k_proto<<<C_, 32, 0, stream>>>(feats, bank, invnorm, owner, cnt_new, proto);
}